// MaskedViterbi_56092272886014
// MI455X (gfx1250) — compile-verified
//
#include <hip/hip_runtime.h>
#include <math.h>

// Problem shape (from reference setup_inputs): B=8, T=512, S=128.
#define S_DIM    128
#define DEPTH    4            // TDM pipeline depth: 4 x 64KB tiles in LDS
#define NTHREADS 1024         // 32 waves of 32 lanes

typedef __attribute__((ext_vector_type(4))) unsigned int v4u;
typedef __attribute__((ext_vector_type(8))) int          v8i;
typedef __attribute__((ext_vector_type(4))) int          v4i;

// Issue one TDM load of a contiguous 64KB (128x128 f32) step-tile into LDS.
// D# per CDNA5 ISA ch.8: group0 = {ctrl, lds_addr, global_addr_lo, global_addr_hi|type},
// group1 = {data_size=4B, tensor_dim0=16384, tensor_dim1=1, tile_dim0=16384,
//           tile_dim1=1, strides=16384}. Groups 2/3 unused (<=2D tensor).
__device__ __forceinline__ void tdm_load_tile(const float* gsrc, unsigned lds_byte_off) {
  unsigned long long ga = (unsigned long long)(size_t)gsrc;
  v4u g0 = {
    1u,                                                   // count=1, user mode
    lds_byte_off,                                         // lds_addr (bytes)
    (unsigned)(ga & 0xffffffffull),                       // global_addr[31:0]
    (unsigned)((ga >> 32) & 0x1ffffffull) | (2u << 30)    // global_addr[56:32] | type=2
  };
  v8i g1 = {
    0x00020000,                 // workgroup_mask=0, data_size=2 (4 bytes)
    0x40000000,                 // tensor_dim0[15:0]=0x4000 (16384) in bits 63:48
    0x00010000,                 // tensor_dim0 hi=0, tensor_dim1=1
    0x40000000,                 // tensor_dim1 hi=0, tile_dim0=16384
    1,                          // tile_dim1=1, tile_dim2=0
    16384,                      // tensor_dim0_stride[31:0]
    0x40000000,                 // stride0 hi=0, tensor_dim1_stride[15:0]=16384
    0                           // tensor_dim1_stride hi
  };
  v4i z4 = {0, 0, 0, 0};
  v8i z8 = {0, 0, 0, 0, 0, 0, 0, 0};
  __builtin_amdgcn_tensor_load_to_lds(g0, g1, z4, z4, z8, 0);
}

__global__ __launch_bounds__(NTHREADS)
void masked_viterbi_scan(const float* __restrict__ theta,
                         const float* __restrict__ mask,
                         float* __restrict__ out,
                         int T) {
  extern __shared__ __align__(16) float smem[];   // DEPTH tiles | V[128] | mask[T]
  float* __restrict__ sV = smem + DEPTH * S_DIM * S_DIM;
  float* __restrict__ sM = sV + S_DIM;

  const int b    = blockIdx.x;
  const int tid  = threadIdx.x;
  const int wave = tid >> 5;
  const int lane = tid & 31;

  const float* __restrict__ thb = theta + (size_t)b * T * S_DIM * S_DIM;
  const float* __restrict__ mb  = mask  + (size_t)b * (T + 1) + 1;  // mask[:,1:]

  if (tid < S_DIM) sV[tid] = 0.0f;                       // V0 = 0
  for (int t = tid; t < T; t += NTHREADS) sM[t] = mb[t]; // stage mask row in LDS

  // Prologue: wave 0 primes the DEPTH-deep TDM pipeline and waits for tile 0.
  if (wave == 0) {
    const int npre = (T < DEPTH) ? T : DEPTH;
    for (int t = 0; t < npre; ++t)
      tdm_load_tile(thb + (size_t)t * S_DIM * S_DIM,
                    (unsigned)((t & (DEPTH - 1)) * S_DIM * S_DIM * sizeof(float)));
    if (npre >= 4)      __builtin_amdgcn_s_wait_tensorcnt(3);
    else if (npre == 3) __builtin_amdgcn_s_wait_tensorcnt(2);
    else if (npre == 2) __builtin_amdgcn_s_wait_tensorcnt(1);
    else if (npre == 1) __builtin_amdgcn_s_wait_tensorcnt(0);
  }
  __syncthreads();                               // (A) tile 0 + sV + sM visible

  for (int t = 0; t < T; ++t) {
    const float* __restrict__ th = smem + (t & (DEPTH - 1)) * S_DIM * S_DIM;
    const float mval = sM[t];

    // Each wave reduces 4 rows; each lane handles 4 contiguous columns.
    const float4 vv = *(const float4*)(sV + lane * 4);
    float vnew[4];
#pragma unroll
    for (int rr = 0; rr < 4; ++rr) {
      const int row = wave + rr * 32;
      const float4 tv = *(const float4*)(th + row * S_DIM + lane * 4);
      const float a0 = tv.x + vv.x, a1 = tv.y + vv.y;
      const float a2 = tv.z + vv.z, a3 = tv.w + vv.w;
      float mx = fmaxf(fmaxf(a0, a1), fmaxf(a2, a3));
#pragma unroll
      for (int off = 16; off > 0; off >>= 1)
        mx = fmaxf(mx, __shfl_xor(mx, off, 32));
      float s = __expf(a0 - mx) + __expf(a1 - mx) + __expf(a2 - mx) + __expf(a3 - mx);
#pragma unroll
      for (int off = 16; off > 0; off >>= 1)
        s += __shfl_xor(s, off, 32);
      vnew[rr] = mx + __logf(s);
    }
    __syncthreads();                             // (B) reads of tile t & sV done

    if (lane == 0) {
#pragma unroll
      for (int rr = 0; rr < 4; ++rr) {
        const int row = wave + rr * 32;
        sV[row] = mval * vnew[rr] + (1.0f - mval) * sV[row];
      }
    }
    // Wave 0: refill the just-freed slot with tile t+DEPTH, then guarantee
    // tile t+1 is resident before releasing barrier (A) of the next step.
    if (wave == 0) {
      if (t + DEPTH < T)
        tdm_load_tile(thb + (size_t)(t + DEPTH) * S_DIM * S_DIM,
                      (unsigned)(((t + DEPTH) & (DEPTH - 1)) * S_DIM * S_DIM * sizeof(float)));
      const int rem = T - 1 - t;                 // tiles still in flight after refill
      if (rem >= DEPTH)   __builtin_amdgcn_s_wait_tensorcnt(3);
      else if (rem == 3)  __builtin_amdgcn_s_wait_tensorcnt(2);
      else if (rem == 2)  __builtin_amdgcn_s_wait_tensorcnt(1);
      else if (rem == 1)  __builtin_amdgcn_s_wait_tensorcnt(0);
      // rem == 0: last iteration, nothing left to wait for
    }
    __syncthreads();                             // (A) tile t+1 ready, sV published
  }

  // Final: out[b] = logsumexp(V_T)
  if (wave == 0) {
    const float4 v = *(const float4*)(sV + lane * 4);
    float mx = fmaxf(fmaxf(v.x, v.y), fmaxf(v.z, v.w));
#pragma unroll
    for (int off = 16; off > 0; off >>= 1)
      mx = fmaxf(mx, __shfl_xor(mx, off, 32));
    float s = __expf(v.x - mx) + __expf(v.y - mx) + __expf(v.z - mx) + __expf(v.w - mx);
#pragma unroll
    for (int off = 16; off > 0; off >>= 1)
      s += __shfl_xor(s, off, 32);
    if (lane == 0) out[b] = mx + __logf(s);
  }
}

extern "C" void kernel_launch(void* const* d_in, const int* in_sizes, int n_in,
                              void* d_out, int out_size, void* d_ws, size_t ws_size,
                              hipStream_t stream) {
  (void)n_in; (void)d_ws; (void)ws_size;
  const float* theta = (const float*)d_in[0];
  const float* mask  = (const float*)d_in[1];
  float* out = (float*)d_out;

  const int B = out_size;                       // 8 batches -> 8 outputs
  const int T = in_sizes[1] / B - 1;            // mask is (B, T+1)

  const size_t shbytes = (size_t)DEPTH * S_DIM * S_DIM * sizeof(float)
                       + (size_t)(S_DIM + T) * sizeof(float);
  (void)hipFuncSetAttribute((const void*)masked_viterbi_scan,
                            hipFuncAttributeMaxDynamicSharedMemorySize,
                            (int)shbytes);
  masked_viterbi_scan<<<dim3(B), dim3(NTHREADS), shbytes, stream>>>(theta, mask, out, T);
}